// ST_GCN_16209206575697
// MI455X (gfx1250) — compile-verified
//
#include <hip/hip_runtime.h>

// ---------------------------------------------------------------------------
// ST-GCN forward for MI455X (gfx1250), wave32, WMMA f32 16x16x4.
// v-major PM layout: [position p][col], col = v*C + c.  All tensors are
// L2-resident (192MB); hot loops are engineered for issue rate:
//   - uniform base-pointer bump per K step + loop-invariant lane offsets
//     (saddr-form b64 loads, no per-iteration VALU address math)
//   - wave-uniform interior/boundary split in the temporal conv (__all vote)
//   - b128 stores in both epilogues; shifts/ands instead of div/mod
// ---------------------------------------------------------------------------

#define VJ   17     // graph vertices
#define TKS  9      // temporal kernel size
#define HOPK 3      // adjacency hops (HOP=2 -> K=3)

typedef __attribute__((ext_vector_type(2))) float v2f;
typedef __attribute__((ext_vector_type(4))) float v4f;
typedef __attribute__((ext_vector_type(8))) float v8f;

__device__ __forceinline__ v8f wmma_f32_16x16x4(v2f a, v2f b, v8f c) {
  return __builtin_amdgcn_wmma_f32_16x16x4_f32(false, a, false, b, (short)0, c,
                                               false, false);
}

// ---------------------------------------------------------------------------
// Deterministic two-stage BN statistics (no float atomics).
// ---------------------------------------------------------------------------

__global__ void stats_pm(const float* __restrict__ X, float* __restrict__ part,
                         int C, int P, int nchunks) {
  const int c = blockIdx.x;
  const int chunk = blockIdx.y;
  const int Mcols = C * VJ;
  float s = 0.f, sq = 0.f;
  for (int p = chunk * blockDim.x + threadIdx.x; p < P;
       p += nchunks * blockDim.x) {
    unsigned base = (unsigned)p * Mcols + c;
#pragma unroll
    for (int v = 0; v < VJ; ++v) {
      float val = X[base + (unsigned)v * C];
      s += val;
      sq += val * val;
    }
  }
  __shared__ float rs[256], rq[256];
  rs[threadIdx.x] = s; rq[threadIdx.x] = sq;
  __syncthreads();
  for (int off = blockDim.x >> 1; off > 0; off >>= 1) {
    if (threadIdx.x < off) {
      rs[threadIdx.x] += rs[threadIdx.x + off];
      rq[threadIdx.x] += rq[threadIdx.x + off];
    }
    __syncthreads();
  }
  if (threadIdx.x == 0) {
    part[(c * nchunks + chunk) * 2 + 0] = rs[0];
    part[(c * nchunks + chunk) * 2 + 1] = rq[0];
  }
}

// bn0 variant: channel vc in [0,51) -> (v = vc/3, c = vc%3) of [N,3,T,V] input.
__global__ void bn0_partial(const float* __restrict__ X,
                            float* __restrict__ part, int T, int N,
                            int nchunks) {
  const int vc = blockIdx.x;
  const int v = vc / 3;
  const int c = vc - v * 3;
  const int chunk = blockIdx.y;
  const int total = N * T;
  float s = 0.f, sq = 0.f;
  for (int i = chunk * blockDim.x + threadIdx.x; i < total;
       i += nchunks * blockDim.x) {
    int n = i / T;
    int t = i - n * T;
    float val = X[(((long)n * 3 + c) * T + t) * VJ + v];
    s += val;
    sq += val * val;
  }
  __shared__ float rs[256], rq[256];
  rs[threadIdx.x] = s; rq[threadIdx.x] = sq;
  __syncthreads();
  for (int off = blockDim.x >> 1; off > 0; off >>= 1) {
    if (threadIdx.x < off) {
      rs[threadIdx.x] += rs[threadIdx.x + off];
      rq[threadIdx.x] += rq[threadIdx.x + off];
    }
    __syncthreads();
  }
  if (threadIdx.x == 0) {
    part[(vc * nchunks + chunk) * 2 + 0] = rs[0];
    part[(vc * nchunks + chunk) * 2 + 1] = rq[0];
  }
}

__global__ void stats_finalize(const float* __restrict__ part, int nchunks,
                               const float* __restrict__ g,
                               const float* __restrict__ b,
                               float* __restrict__ scale,
                               float* __restrict__ shift, float cnt) {
  const int c = blockIdx.x;
  float s = 0.f, sq = 0.f;
  for (int i = threadIdx.x; i < nchunks; i += blockDim.x) {
    s += part[(c * nchunks + i) * 2 + 0];
    sq += part[(c * nchunks + i) * 2 + 1];
  }
  __shared__ float rs[256], rq[256];
  rs[threadIdx.x] = s; rq[threadIdx.x] = sq;
  __syncthreads();
  for (int off = blockDim.x >> 1; off > 0; off >>= 1) {
    if (threadIdx.x < off) {
      rs[threadIdx.x] += rs[threadIdx.x + off];
      rq[threadIdx.x] += rq[threadIdx.x + off];
    }
    __syncthreads();
  }
  if (threadIdx.x == 0) {
    float mean = rs[0] / cnt;
    float var = rq[0] / cnt - mean * mean;
    float istd = rsqrtf(var + 1e-5f);
    float sc = g[c] * istd;
    scale[c] = sc;
    shift[c] = b[c] - mean * sc;
  }
}

// In-place BN+ReLU over PM layout; one block per position row; c = col & Cmask.
__global__ void bn_apply_pm(float* __restrict__ X,
                            const float* __restrict__ scale,
                            const float* __restrict__ shift, int Mcols,
                            int Cmask) {
  unsigned base = (unsigned)blockIdx.x * Mcols;
  for (int col = threadIdx.x; col < Mcols; col += blockDim.x) {
    int c = col & Cmask;
    X[base + col] = fmaxf(0.f, X[base + col] * scale[c] + shift[c]);
  }
}

// bn0 apply + transpose [N,3,T,V] -> PM [(n*T+t)][v*3+c]  (no ReLU).
__global__ void bn0_apply_pm(const float* __restrict__ X,
                             const float* __restrict__ scale0,
                             const float* __restrict__ shift0,
                             float* __restrict__ H, int T, int N) {
  const int Mcols = 3 * VJ;  // 51
  const long total = (long)N * T * Mcols;
  for (long i = blockIdx.x * (long)blockDim.x + threadIdx.x; i < total;
       i += (long)gridDim.x * blockDim.x) {
    long p = i / Mcols;                    // n*T + t
    int col = (int)(i - p * Mcols);        // v*3 + c
    int v = col / 3;
    int c = col - v * 3;
    int n = (int)(p / T);
    int t = (int)(p - (long)n * T);
    float val = X[(((long)n * 3 + c) * T + t) * VJ + v];
    H[i] = val * scale0[col] + shift0[col];
  }
}

// ---------------------------------------------------------------------------
// Fused spatial-conv + graph weights (v-major layout):
//   row m = w*cout + c, col k = v*cin + ci
// ---------------------------------------------------------------------------
__global__ void weff_build(const float* __restrict__ ws_,
                           const float* __restrict__ A,
                           const float* __restrict__ M,
                           float* __restrict__ Weff, int cin, int cout) {
  const int Kd = cin * VJ;
  const long total = (long)cout * VJ * Kd;
  for (long i = blockIdx.x * (long)blockDim.x + threadIdx.x; i < total;
       i += (long)gridDim.x * blockDim.x) {
    int row = (int)(i / Kd);
    int col = (int)(i - (long)row * Kd);
    int v = col / cin;
    int ci = col - v * cin;
    int w = row / cout;
    int c = row - w * cout;
    float acc = 0.f;
#pragma unroll
    for (int h = 0; h < HOPK; ++h) {
      int avw = (h * VJ + v) * VJ + w;
      acc += ws_[((long)h * cout + c) * cin + ci] * A[avw] * M[avw];
    }
    Weff[i] = acc;
  }
}

__global__ void beff_build(const float* __restrict__ bs,
                           const float* __restrict__ A,
                           const float* __restrict__ M,
                           float* __restrict__ beff, int cout) {
  int i = blockIdx.x * blockDim.x + threadIdx.x;
  if (i >= cout * VJ) return;
  int w = i / cout;
  int c = i - w * cout;
  float acc = 0.f;
  for (int h = 0; h < HOPK; ++h) {
    float colsum = 0.f;
#pragma unroll
    for (int v = 0; v < VJ; ++v) {
      int avw = (h * VJ + v) * VJ + w;
      colsum += A[avw] * M[avw];
    }
    acc += bs[h * cout + c] * colsum;
  }
  beff[i] = acc;
}

// Permute temporal weights [co][ci][dt] -> [co][dt*cin + ci].
__global__ void wt_perm(const float* __restrict__ wt, float* __restrict__ wtp,
                        int cin, int cout) {
  const int Kd = cin * TKS;
  int i = blockIdx.x * blockDim.x + threadIdx.x;
  if (i >= cout * Kd) return;
  int co = i / Kd;
  int k = i - co * Kd;
  int dt = k / cin;
  int ci = k - dt * cin;
  wtp[i] = wt[((long)co * cin + ci) * TKS + dt];
}

// ---------------------------------------------------------------------------
// Spatial+graph GEMM: Y[p][m] = sum_k Weff[m][k]*X[p][k] + beff[m].
// 16(M) x 64(P) per wave.  Unguarded path: uniform base bump + invariant lane
// offsets -> pure b64 saddr loads, zero address VALU per iteration.
// ---------------------------------------------------------------------------
template <bool GUARD>
__global__ void __launch_bounds__(128) spatial_wmma(
    const float* __restrict__ X,     // [P][Kd]
    const float* __restrict__ Weff,  // [Mrows][Kd]
    const float* __restrict__ beff,  // [Mrows]
    float* __restrict__ Y,           // [P][Mrows]
    int Kd, int Mrows, int numM, int nP64) {
  const int lane = threadIdx.x & 31;
  const int wv = threadIdx.x >> 5;
  const int tile = blockIdx.x * 4 + wv;
  if (tile >= numM * nP64) return;      // wave-uniform
  const int mt = tile % numM;
  const int p0 = (tile / numM) * 64;
  const int l15 = lane & 15;
  const int klo = (lane >> 4) << 1;     // 0 or 2

  v8f acc[4] = {{}, {}, {}, {}};
  if (GUARD) {
    const unsigned wb = (unsigned)(mt * 16 + l15) * Kd;
    unsigned pb[4];
#pragma unroll
    for (int q = 0; q < 4; ++q) pb[q] = (unsigned)(p0 + q * 16 + l15) * Kd;
    for (int kk = 0; kk < Kd; kk += 4) {
      int k0 = kk + klo, k1 = k0 + 1;
      unsigned k0c = k0 < Kd ? (unsigned)k0 : 0u;
      unsigned k1c = k1 < Kd ? (unsigned)k1 : 0u;
      float m0 = k0 < Kd ? 1.f : 0.f;
      float m1 = k1 < Kd ? 1.f : 0.f;
      v2f a;
      a.x = Weff[wb + k0c] * m0;
      a.y = Weff[wb + k1c] * m1;
#pragma unroll
      for (int q = 0; q < 4; ++q) {
        v2f b;
        b.x = X[pb[q] + k0c] * m0;
        b.y = X[pb[q] + k1c] * m1;
        acc[q] = wmma_f32_16x16x4(a, b, acc[q]);
      }
    }
  } else {
    // loop-invariant lane offsets (even -> 8B aligned)
    const unsigned cw = (unsigned)(mt * 16 + l15) * Kd + klo;
    unsigned cq[4];
#pragma unroll
    for (int q = 0; q < 4; ++q)
      cq[q] = (unsigned)(p0 + q * 16 + l15) * Kd + klo;
    for (int kk = 0; kk < Kd; kk += 4) {
      const float* Wk = Weff + kk;      // uniform base bump
      const float* Xk = X + kk;
      v2f a = *(const v2f*)(Wk + cw);
#pragma unroll
      for (int q = 0; q < 4; ++q) {
        v2f b = *(const v2f*)(Xk + cq[q]);
        acc[q] = wmma_f32_16x16x4(a, b, acc[q]);
      }
    }
  }

  const int mbase = mt * 16 + ((lane >> 4) << 3);
  const v4f bb0 = *(const v4f*)(beff + mbase);
  const v4f bb1 = *(const v4f*)(beff + mbase + 4);
#pragma unroll
  for (int q = 0; q < 4; ++q) {
    unsigned base = (unsigned)(p0 + q * 16 + l15) * Mrows + mbase;
    v4f lo, hi;
    lo.x = acc[q][0] + bb0.x; lo.y = acc[q][1] + bb0.y;
    lo.z = acc[q][2] + bb0.z; lo.w = acc[q][3] + bb0.w;
    hi.x = acc[q][4] + bb1.x; hi.y = acc[q][5] + bb1.y;
    hi.z = acc[q][6] + bb1.z; hi.w = acc[q][7] + bb1.w;
    *(v4f*)(Y + base) = lo;
    *(v4f*)(Y + base + 4) = hi;
  }
}

// ---------------------------------------------------------------------------
// Temporal 9x1 conv, implicit GEMM.  K order = dt*cin + ci (permuted weights).
// The K-step address split  s(kk) = dt*Min + ci_scalar  is wave-uniform, so
// interior tiles use uniform-base + invariant-offset b64 loads; only tiles
// touching the t-padding (selected by a wave-uniform __all vote) take the
// masked path.  cin, Tout are powers of two.
// ---------------------------------------------------------------------------
__global__ void __launch_bounds__(128) temporal_wmma(
    const float* __restrict__ X,    // [n*T+t][v*cin+ci]
    const float* __restrict__ Wtp,  // [cout][dt*cin+ci]
    const float* __restrict__ bt,   // [cout]
    float* __restrict__ Y,          // [n*Tout+to][v*cout+co]
    int cinSh, int cout, int T, int toutSh, int stride, int numM, int nP64) {
  const int cin = 1 << cinSh;
  const int Tout = 1 << toutSh;
  const int lane = threadIdx.x & 31;
  const int wv = threadIdx.x >> 5;
  const int tile = blockIdx.x * 4 + wv;
  if (tile >= numM * nP64) return;
  const int mt = tile % numM;
  const int p0 = (tile / numM) * 64;
  const int l15 = lane & 15;
  const int klo = (lane >> 4) << 1;
  const int Kd = cin * TKS;
  const int Min = cin * VJ;
  const int Mout = cout * VJ;
  const int mbase = mt * 16 + ((lane >> 4) << 3);

  const unsigned cwt = (unsigned)(mt * 16 + l15) * Kd + klo;
  unsigned base[4], obase0[4];
  int tin0[4];
  int ok = 1;
#pragma unroll
  for (int q = 0; q < 4; ++q) {
    int p = p0 + q * 16 + l15;
    int v = p % VJ;
    int r = p / VJ;                      // n*Tout + to
    int to = r & (Tout - 1);
    int n = r >> toutSh;
    base[q] = (unsigned)n * T * Min + (unsigned)v * cin + klo;
    tin0[q] = to * stride - (TKS - 1) / 2;
    obase0[q] = (unsigned)r * Mout + (unsigned)v * cout + mbase;
    ok &= (tin0[q] >= 0) & (tin0[q] + TKS <= T);
  }

  v8f acc[4] = {{}, {}, {}, {}};
  if (__all(ok)) {
    // interior: no boundary taps anywhere in the tile
    unsigned cq[4];
#pragma unroll
    for (int q = 0; q < 4; ++q) cq[q] = base[q] + (unsigned)tin0[q] * Min;
    for (int kk = 0; kk < Kd; kk += 4) {
      int s = (kk >> cinSh) * Min + (kk & (cin - 1));  // uniform
      const float* Xk = X + s;
      const float* Wk = Wtp + kk;
      v2f a = *(const v2f*)(Wk + cwt);
#pragma unroll
      for (int q = 0; q < 4; ++q) {
        v2f b = *(const v2f*)(Xk + cq[q]);
        acc[q] = wmma_f32_16x16x4(a, b, acc[q]);
      }
    }
  } else {
    // boundary: clamp + mask (zero padding)
    for (int kk = 0; kk < Kd; kk += 4) {
      int dt = kk >> cinSh;              // uniform (klo can't cross cin)
      int cis = kk & (cin - 1);          // uniform scalar part of ci
      const float* Wk = Wtp + kk;
      v2f a = *(const v2f*)(Wk + cwt);
#pragma unroll
      for (int q = 0; q < 4; ++q) {
        int t0 = tin0[q] + dt;
        int in0 = ((unsigned)t0 < (unsigned)T);
        unsigned t0c = in0 ? (unsigned)t0 : 0u;
        float msk = in0 ? 1.f : 0.f;
        v2f b = *(const v2f*)(X + base[q] + t0c * Min + cis);
        b.x *= msk;
        b.y *= msk;
        acc[q] = wmma_f32_16x16x4(a, b, acc[q]);
      }
    }
  }

  const v4f bb0 = *(const v4f*)(bt + mbase);
  const v4f bb1 = *(const v4f*)(bt + mbase + 4);
#pragma unroll
  for (int q = 0; q < 4; ++q) {
    v4f lo, hi;
    lo.x = acc[q][0] + bb0.x; lo.y = acc[q][1] + bb0.y;
    lo.z = acc[q][2] + bb0.z; lo.w = acc[q][3] + bb0.w;
    hi.x = acc[q][4] + bb1.x; hi.y = acc[q][5] + bb1.y;
    hi.z = acc[q][6] + bb1.z; hi.w = acc[q][7] + bb1.w;
    *(v4f*)(Y + obase0[q]) = lo;
    *(v4f*)(Y + obase0[q] + 4) = hi;
  }
}

// ---------------------------------------------------------------------------
// Global average pool with fused final BN2+ReLU (v-major PM), then FC.
// ---------------------------------------------------------------------------
__global__ void pool_kernel(const float* __restrict__ X,
                            const float* __restrict__ scale,
                            const float* __restrict__ shift,
                            float* __restrict__ pooled, int C, int Tout) {
  const int nc = blockIdx.x;
  const int n = nc / C;
  const int c = nc - n * C;
  const int Mcols = C * VJ;
  const float sc = scale[c], sh = shift[c];
  float s = 0.f;
  for (int i = threadIdx.x; i < Tout * VJ; i += blockDim.x) {
    int to = i / VJ;
    int v = i - to * VJ;
    float val = X[(unsigned)(n * Tout + to) * Mcols + (unsigned)v * C + c];
    s += fmaxf(0.f, val * sc + sh);
  }
  __shared__ float rs[256];
  rs[threadIdx.x] = s;
  __syncthreads();
  for (int off = blockDim.x >> 1; off > 0; off >>= 1) {
    if (threadIdx.x < off) rs[threadIdx.x] += rs[threadIdx.x + off];
    __syncthreads();
  }
  if (threadIdx.x == 0) pooled[nc] = rs[0] / (float)(Tout * VJ);
}

__global__ void fc_kernel(const float* __restrict__ pooled,
                          const float* __restrict__ fcw,
                          const float* __restrict__ fcb,
                          float* __restrict__ out, int C, int NCLS, int N) {
  int i = blockIdx.x * blockDim.x + threadIdx.x;
  if (i >= N * NCLS) return;
  int n = i / NCLS;
  int k = i - n * NCLS;
  float s = fcb[k];
  for (int c = 0; c < C; ++c) s += pooled[n * C + c] * fcw[k * C + c];
  out[i] = s;
}

// ---------------------------------------------------------------------------
// Host-side launch orchestration.
// ---------------------------------------------------------------------------
extern "C" void kernel_launch(void* const* d_in, const int* in_sizes, int n_in,
                              void* d_out, int out_size, void* d_ws,
                              size_t ws_size, hipStream_t stream) {
  (void)in_sizes; (void)n_in; (void)out_size; (void)ws_size;
  const int N = 64, NCLS = 60, NB = 128;
  const int cinA[6]  = {3, 32, 32, 32, 64, 64};
  const int coutA[6] = {32, 32, 32, 64, 64, 64};
  const int strA[6]  = {1, 1, 1, 2, 1, 1};
  const int TinA[6]  = {512, 512, 512, 512, 256, 256};

  // ---- unpack inputs (setup_inputs() dict insertion order, recursive) ----
  int ix = 0;
  const float* x = (const float*)d_in[ix++];
  const float* bn0_g = (const float*)d_in[ix++];
  const float* bn0_b = (const float*)d_in[ix++];
  const float *w_s[6], *b_s[6], *Mm[6], *bn1g[6], *bn1b[6], *w_t[6], *b_t[6],
      *bn2g[6], *bn2b[6];
  for (int b = 0; b < 6; ++b) {
    w_s[b]  = (const float*)d_in[ix++];
    b_s[b]  = (const float*)d_in[ix++];
    Mm[b]   = (const float*)d_in[ix++];
    bn1g[b] = (const float*)d_in[ix++];
    bn1b[b] = (const float*)d_in[ix++];
    w_t[b]  = (const float*)d_in[ix++];
    b_t[b]  = (const float*)d_in[ix++];
    bn2g[b] = (const float*)d_in[ix++];
    bn2b[b] = (const float*)d_in[ix++];
  }
  const float* fc_w = (const float*)d_in[ix++];
  const float* fc_b = (const float*)d_in[ix++];
  const float* Aadj = (const float*)d_in[ix++];

  // ---- workspace carve-out ----
  char* ws = (char*)d_ws;
  size_t off = 0;
  auto carve = [&](size_t bytes) {
    size_t o = off;
    off += (bytes + 255) & ~(size_t)255;
    return (float*)(ws + o);
  };
  float* buf1   = carve((size_t)64 * 512 * 64 * VJ * 4);  // [P][cout*17] max
  float* bufX   = carve((size_t)64 * 512 * 32 * VJ * 4);  // ping
  float* bufY   = carve((size_t)64 * 512 * 32 * VJ * 4);  // pong
  float* weff   = carve((size_t)64 * VJ * 64 * VJ * 4);
  float* beff   = carve((size_t)64 * VJ * 4);
  float* wtp    = carve((size_t)64 * 64 * TKS * 4);
  float* part   = carve((size_t)64 * NB * 2 * 4);
  float* scale0 = carve(64 * 4);
  float* shift0 = carve(64 * 4);
  float* scale1 = carve(64 * 4);
  float* shift1 = carve(64 * 4);
  float* scale2 = carve(64 * 4);
  float* shift2 = carve(64 * 4);
  float* pooled = carve((size_t)64 * 64 * 4);

  // ---- data BN (bn0) + transpose to PM layout ----
  bn0_partial<<<dim3(VJ * 3, NB), 256, 0, stream>>>(x, part, 512, N, NB);
  stats_finalize<<<VJ * 3, 256, 0, stream>>>(part, NB, bn0_g, bn0_b, scale0,
                                             shift0, (float)(N * 512));
  bn0_apply_pm<<<2048, 256, 0, stream>>>(x, scale0, shift0, bufX, 512, N);

  // ---- STGC blocks ----
  float* cur = bufX;
  float* nxt = bufY;
  for (int b = 0; b < 6; ++b) {
    const int cin = cinA[b], cout = coutA[b], T = TinA[b], stride = strA[b];
    const int Tout = T / stride;
    const int P = N * T;
    const int Kd = cin * VJ;
    const int Mrows = cout * VJ;

    // normalize the block input in place (previous block's BN2 + ReLU)
    if (b > 0)
      bn_apply_pm<<<P, 256, 0, stream>>>(cur, scale2, shift2, Kd, cin - 1);

    weff_build<<<2048, 256, 0, stream>>>(w_s[b], Aadj, Mm[b], weff, cin, cout);
    beff_build<<<(cout * VJ + 255) / 256, 256, 0, stream>>>(b_s[b], Aadj,
                                                            Mm[b], beff, cout);
    wt_perm<<<(cout * cout * TKS + 255) / 256, 256, 0, stream>>>(w_t[b], wtp,
                                                                 cout, cout);

    // fused spatial conv + graph contraction
    const int numM = Mrows / 16;
    const int nP64 = P / 64;
    const long tiles = (long)numM * nP64;
    if (b == 0)
      spatial_wmma<true><<<(int)((tiles + 3) / 4), 128, 0, stream>>>(
          cur, weff, beff, buf1, Kd, Mrows, numM, nP64);
    else
      spatial_wmma<false><<<(int)((tiles + 3) / 4), 128, 0, stream>>>(
          cur, weff, beff, buf1, Kd, Mrows, numM, nP64);

    // BN1 stats, then normalize buf1 in place (BN1 + ReLU)
    stats_pm<<<dim3(cout, NB), 256, 0, stream>>>(buf1, part, cout, P, NB);
    stats_finalize<<<cout, 256, 0, stream>>>(part, NB, bn1g[b], bn1b[b],
                                             scale1, shift1,
                                             (float)((long)P * VJ));
    bn_apply_pm<<<P, 256, 0, stream>>>(buf1, scale1, shift1, Mrows, cout - 1);

    // temporal conv (cout, Tout are powers of two)
    const int numM2 = cout / 16;
    const int nP64_2 = N * Tout * VJ / 64;
    const long tiles2 = (long)numM2 * nP64_2;
    temporal_wmma<<<(int)((tiles2 + 3) / 4), 128, 0, stream>>>(
        buf1, wtp, b_t[b], nxt, __builtin_ctz(cout), cout, T,
        __builtin_ctz(Tout), stride, numM2, nP64_2);

    // BN2 stats (applied lazily by the next block / the pool)
    stats_pm<<<dim3(cout, NB), 256, 0, stream>>>(nxt, part, cout, N * Tout,
                                                 NB);
    stats_finalize<<<cout, 256, 0, stream>>>(part, NB, bn2g[b], bn2b[b],
                                             scale2, shift2,
                                             (float)((long)N * Tout * VJ));

    float* tmp = cur; cur = nxt; nxt = tmp;
  }

  // ---- pool (fused final BN2+ReLU) + FC ----
  pool_kernel<<<N * 64, 256, 0, stream>>>(cur, scale2, shift2, pooled, 64,
                                          256);
  fc_kernel<<<(N * NCLS + 255) / 256, 256, 0, stream>>>(
      pooled, fc_w, fc_b, (float*)d_out, 64, NCLS, N);
}